// Model_SemSeg_35931696398579
// MI455X (gfx1250) — compile-verified
//
#include <hip/hip_runtime.h>
#include <math.h>

typedef float v2f __attribute__((ext_vector_type(2)));
typedef float v8f __attribute__((ext_vector_type(8)));

#define BATCH 8
#define NPTS  4096
#define KNN   20
#define NK    (NPTS * KNN)   // 81920

// ---------------------------------------------------------------------------
// kNN: one block per (point i, batch b). Distances to all j in LDS, then
// KNN selection passes (arg-min with index tie-break, self excluded).
// ---------------------------------------------------------------------------
template<int D>
__global__ __launch_bounds__(256)
void knn_kernel(const float* __restrict__ xs, long bstride, int cstride,
                int* __restrict__ idx_out)
{
    const int i = blockIdx.x;
    const int b = blockIdx.y;
    const int tid = threadIdx.x;
    const float* xb = xs + (size_t)b * bstride;

    __shared__ float xi[D];
    __shared__ float dist[NPTS];
    __shared__ float rv[256];
    __shared__ int   ri[256];

    for (int c = tid; c < D; c += 256) xi[c] = xb[(size_t)c * cstride + i];
    __syncthreads();

    for (int j = tid; j < NPTS; j += 256) {
        float s = 0.f;
#pragma unroll 8
        for (int c = 0; c < D; ++c) {
            float d = xb[(size_t)c * cstride + j] - xi[c];
            s += d * d;
        }
        dist[j] = (j == i) ? 3.4e38f : s;
    }
    __syncthreads();

    for (int r = 0; r < KNN; ++r) {
        float best = 3.4e38f; int bi = 0x7fffffff;
        for (int j = tid; j < NPTS; j += 256) {
            float v = dist[j];
            if (v < best || (v == best && j < bi)) { best = v; bi = j; }
        }
        rv[tid] = best; ri[tid] = bi;
        __syncthreads();
        for (int st = 128; st > 0; st >>= 1) {
            if (tid < st) {
                float v2 = rv[tid + st]; int i2 = ri[tid + st];
                if (v2 < rv[tid] || (v2 == rv[tid] && i2 < ri[tid])) {
                    rv[tid] = v2; ri[tid] = i2;
                }
            }
            __syncthreads();
        }
        int w = ri[0];
        if (tid == 0) {
            idx_out[((size_t)b * NPTS + i) * KNN + r] = w;
            dist[w] = 3.4e38f;
        }
        __syncthreads();
    }
}

// ---------------------------------------------------------------------------
// Edge-conv 1 (Cin=3 -> 6 edge channels, W1 64x6), fused gather.
// One wave per point: lane = neighbor slot (kk < 20), loop over o.
// Output layout (b, 64, NK) with m = i*20+kk.
// ---------------------------------------------------------------------------
__global__ __launch_bounds__(256)
void edgeconv1_kernel(const float* __restrict__ x, const int* __restrict__ idx,
                      const float* __restrict__ W1, float* __restrict__ A)
{
    const int g    = blockIdx.x * 8 + (threadIdx.x >> 5);   // global point, b*n total
    const int lane = threadIdx.x & 31;
    const int b = g / NPTS;
    const int i = g - b * NPTS;
    const float* xb = x + (size_t)b * 3 * NPTS;

    const float xi0 = xb[i], xi1 = xb[NPTS + i], xi2 = xb[2 * NPTS + i];
    float d0 = 0.f, d1 = 0.f, d2 = 0.f;
    const bool active = lane < KNN;
    if (active) {
        int j = idx[(size_t)g * KNN + lane];
        d0 = xb[j] - xi0;
        d1 = xb[NPTS + j] - xi1;
        d2 = xb[2 * NPTS + j] - xi2;
    }
    float* Ab = A + (size_t)b * 64 * NK + (size_t)i * KNN + lane;
    for (int o = 0; o < 64; ++o) {
        const float* w = W1 + o * 6;
        float v = d0 * w[0] + d1 * w[1] + d2 * w[2]
                + xi0 * w[3] + xi1 * w[4] + xi2 * w[5];
        if (active) Ab[(size_t)o * NK] = v;
    }
}

// ---------------------------------------------------------------------------
// WMMA f32 GEMM:  Y[b][o][m] = sum_c W[o][c] * X[b][c][m]
// EDGE variant synthesizes X on the fly as graph_feature(xsrc, idx):
//   c <  64 : xsrc[c][j(m)] - xsrc[c][i(m)]
//   c >= 64 : xsrc[c-64][i(m)]
// Block = 256 threads = 8 waves; wave tile = 16(O) x 64(M); block = 64(O) x 128(M).
// Uses V_WMMA_F32_16X16X4_F32 (exact f32 matrix op).
// ---------------------------------------------------------------------------
template<bool EDGE>
__global__ __launch_bounds__(256)
void gemm_wmma(const float* __restrict__ W, const float* __restrict__ X,
               float* __restrict__ Y, int O, int Cin, long Mp,
               const float* __restrict__ xsrc, long xbst, int xcst,
               const int* __restrict__ idx)
{
    const int lane  = threadIdx.x & 31;
    const int wave  = threadIdx.x >> 5;
    const int waveO = wave & 3;
    const int waveM = wave >> 2;
    const int b     = blockIdx.z;
    const int obase = blockIdx.y * 64 + waveO * 16;
    const long mbase = (long)blockIdx.x * 128 + waveM * 64;

    const int ln15  = lane & 15;
    const int khalf = (lane >> 4) << 1;      // 0 or 2: this lane's K sub-pair

    long col[4]; int ei[4], ej[4];
#pragma unroll
    for (int t = 0; t < 4; ++t) {
        col[t] = mbase + t * 16 + ln15;
        if (EDGE) {
            long ip = col[t] / KNN;
            int  kk = (int)(col[t] - ip * KNN);
            ei[t] = (int)ip;
            ej[t] = idx[((long)b * NPTS + ip) * KNN + kk];
        }
    }

    const float* Wb = W + (size_t)(obase + ln15) * Cin;
    const float* Xb = EDGE ? (xsrc + (size_t)b * xbst)
                           : (X + (size_t)b * Cin * Mp);

    v8f acc[4];
#pragma unroll
    for (int t = 0; t < 4; ++t)
#pragma unroll
        for (int r = 0; r < 8; ++r) acc[t][r] = 0.f;

    for (int k0 = 0; k0 < Cin; k0 += 4) {
        const int c0 = k0 + khalf;
        v2f a;
        a.x = Wb[c0];
        a.y = Wb[c0 + 1];

        v2f bt[4];
        if constexpr (!EDGE) {
            const float* p0 = Xb + (size_t)c0 * Mp;
            const float* p1 = p0 + Mp;
#pragma unroll
            for (int t = 0; t < 4; ++t) { bt[t].x = p0[col[t]]; bt[t].y = p1[col[t]]; }
        } else {
            if (k0 < 64) {   // uniform per iteration (k0 multiple of 4)
                const float* p0 = Xb + (size_t)c0 * xcst;
                const float* p1 = p0 + xcst;
#pragma unroll
                for (int t = 0; t < 4; ++t) {
                    bt[t].x = p0[ej[t]] - p0[ei[t]];
                    bt[t].y = p1[ej[t]] - p1[ei[t]];
                }
            } else {
                const float* p0 = Xb + (size_t)(c0 - 64) * xcst;
                const float* p1 = p0 + xcst;
#pragma unroll
                for (int t = 0; t < 4; ++t) { bt[t].x = p0[ei[t]]; bt[t].y = p1[ei[t]]; }
            }
        }
#pragma unroll
        for (int t = 0; t < 4; ++t)
            acc[t] = __builtin_amdgcn_wmma_f32_16x16x4_f32(
                false, a, false, bt[t], (short)0, acc[t], false, false);
    }

    float* Yb = Y + ((size_t)b * O + obase) * Mp;
    const int rowoff = (lane >> 4) << 3;
#pragma unroll
    for (int t = 0; t < 4; ++t)
#pragma unroll
        for (int r = 0; r < 8; ++r)
            Yb[(size_t)(r + rowoff) * Mp + col[t]] = acc[t][r];
}

// ---------------------------------------------------------------------------
// BatchNorm (batch stats) + Mish
// ---------------------------------------------------------------------------
__global__ void zero_doubles(double* p, int n)
{
    int t = blockIdx.x * 256 + threadIdx.x;
    if (t < n) p[t] = 0.0;
}

__global__ __launch_bounds__(256)
void bn_stats(const float* __restrict__ Y, int C, long Mp, double* __restrict__ stats)
{
    const int c = blockIdx.y;
    long m0 = (long)blockIdx.x << 16;
    long m1 = m0 + 65536; if (m1 > Mp) m1 = Mp;

    float s = 0.f, sq = 0.f;
    for (int bb = 0; bb < BATCH; ++bb) {
        const float* p = Y + ((size_t)bb * C + c) * Mp;
        for (long m = m0 + threadIdx.x; m < m1; m += 256) {
            float v = p[m]; s += v; sq += v * v;
        }
    }
    __shared__ float ss[256], sqs[256];
    ss[threadIdx.x] = s; sqs[threadIdx.x] = sq;
    __syncthreads();
    for (int st = 128; st > 0; st >>= 1) {
        if (threadIdx.x < st) {
            ss[threadIdx.x]  += ss[threadIdx.x + st];
            sqs[threadIdx.x] += sqs[threadIdx.x + st];
        }
        __syncthreads();
    }
    if (threadIdx.x == 0) {
        atomicAdd(&stats[c],     (double)ss[0]);
        atomicAdd(&stats[C + c], (double)sqs[0]);
    }
}

__global__ void bn_finalize(const double* __restrict__ stats,
                            float* __restrict__ mean, float* __restrict__ rstd,
                            int C, double total)
{
    int c = blockIdx.x * 256 + threadIdx.x;
    if (c >= C) return;
    double m = stats[c] / total;
    double v = stats[C + c] / total - m * m;
    if (v < 0.0) v = 0.0;
    mean[c] = (float)m;
    rstd[c] = (float)(1.0 / sqrt(v + 1e-5));
}

__global__ __launch_bounds__(256)
void bn_mish_apply(float* __restrict__ Y, const float* __restrict__ mean,
                   const float* __restrict__ rstd, int C, long Mp, long total)
{
    long t = (long)blockIdx.x * 256 + threadIdx.x;
    if (t >= total) return;
    int c = (int)((t / Mp) % C);
    float v = (Y[t] - mean[c]) * rstd[c];
    float sp = (v > 20.f) ? v : log1pf(expf(v));
    Y[t] = v * tanhf(sp);
}

// ---------------------------------------------------------------------------
// Reductions / concats / final head
// ---------------------------------------------------------------------------
__global__ void max_k_kernel(const float* __restrict__ Y, float* __restrict__ xcat,
                             int cOff)
{
    long t = (long)blockIdx.x * 256 + threadIdx.x;     // over b*64*n
    if (t >= (long)BATCH * 64 * NPTS) return;
    int i = (int)(t % NPTS);
    int c = (int)((t / NPTS) % 64);
    int b = (int)(t / (64L * NPTS));
    const float* p = Y + ((size_t)b * 64 + c) * NK + (size_t)i * KNN;
    float m = p[0];
#pragma unroll
    for (int q = 1; q < KNN; ++q) m = fmaxf(m, p[q]);
    xcat[((size_t)b * 192 + cOff + c) * NPTS + i] = m;
}

__global__ __launch_bounds__(256)
void max_n_kernel(const float* __restrict__ g, float* __restrict__ gmax)
{
    const int bc = blockIdx.x;                 // b*1024 + c
    const float* p = g + (size_t)bc * NPTS;
    float m = -3.4e38f;
    for (int i = threadIdx.x; i < NPTS; i += 256) m = fmaxf(m, p[i]);
    __shared__ float sm[256];
    sm[threadIdx.x] = m;
    __syncthreads();
    for (int st = 128; st > 0; st >>= 1) {
        if (threadIdx.x < st) sm[threadIdx.x] = fmaxf(sm[threadIdx.x], sm[threadIdx.x + st]);
        __syncthreads();
    }
    if (threadIdx.x == 0) gmax[bc] = sm[0];
}

__global__ void build_cat2(const float* __restrict__ gmax,
                           const float* __restrict__ xcat, float* __restrict__ cat2)
{
    long t = (long)blockIdx.x * 256 + threadIdx.x;     // over b*1216*n
    if (t >= (long)BATCH * 1216 * NPTS) return;
    int i = (int)(t % NPTS);
    int c = (int)((t / NPTS) % 1216);
    int b = (int)(t / (1216L * NPTS));
    float v = (c < 1024) ? gmax[b * 1024 + c]
                         : xcat[((size_t)b * 192 + (c - 1024)) * NPTS + i];
    cat2[t] = v;
}

__global__ void conv_last(const float* __restrict__ W9, const float* __restrict__ y8,
                          float* __restrict__ out)
{
    long t = (long)blockIdx.x * 256 + threadIdx.x;     // over b*13*n
    if (t >= (long)BATCH * 13 * NPTS) return;
    int i = (int)(t % NPTS);
    int o = (int)((t / NPTS) % 13);
    int b = (int)(t / (13L * NPTS));
    const float* p = y8 + (size_t)b * 256 * NPTS;
    float acc = 0.f;
    for (int c = 0; c < 256; ++c) acc += W9[o * 256 + c] * p[(size_t)c * NPTS + i];
    out[t] = acc;
}

// ---------------------------------------------------------------------------
// Host-side orchestration
// ---------------------------------------------------------------------------
static void run_bn_mish(float* Y, int C, long Mp, double* statsd,
                        float* meanf, float* rstdf, hipStream_t s)
{
    zero_doubles<<<(2 * C + 255) / 256, 256, 0, s>>>(statsd, 2 * C);
    int chunks = (int)((Mp + 65535) / 65536);
    bn_stats<<<dim3(chunks, C), 256, 0, s>>>(Y, C, Mp, statsd);
    bn_finalize<<<(C + 255) / 256, 256, 0, s>>>(statsd, meanf, rstdf, C,
                                                (double)BATCH * (double)Mp);
    long tot = (long)BATCH * C * Mp;
    bn_mish_apply<<<(unsigned)((tot + 255) / 256), 256, 0, s>>>(Y, meanf, rstdf, C, Mp, tot);
}

static void run_gemm(const float* W, const float* X, float* Y, int O, int Cin,
                     long Mp, hipStream_t s)
{
    dim3 g((unsigned)(Mp / 128), O / 64, BATCH);
    gemm_wmma<false><<<g, 256, 0, s>>>(W, X, Y, O, Cin, Mp, nullptr, 0, 0, nullptr);
}

static void run_gemm_edge(const float* W, const float* xsrc, long xbst, int xcst,
                          const int* idx, float* Y, int O, int Cin, long Mp,
                          hipStream_t s)
{
    dim3 g((unsigned)(Mp / 128), O / 64, BATCH);
    gemm_wmma<true><<<g, 256, 0, s>>>(W, nullptr, Y, O, Cin, Mp, xsrc, xbst, xcst, idx);
}

extern "C" void kernel_launch(void* const* d_in, const int* in_sizes, int n_in,
                              void* d_out, int out_size, void* d_ws, size_t ws_size,
                              hipStream_t stream)
{
    const float* x  = (const float*)d_in[0];
    const float* W1 = (const float*)d_in[1];
    const float* W2 = (const float*)d_in[2];
    const float* W3 = (const float*)d_in[3];
    const float* W4 = (const float*)d_in[4];
    const float* W5 = (const float*)d_in[5];
    const float* W6 = (const float*)d_in[6];
    const float* W7 = (const float*)d_in[7];
    const float* W8 = (const float*)d_in[8];
    const float* W9 = (const float*)d_in[9];

    float* ws = (float*)d_ws;

    // workspace layout (units: floats)
    const size_t OFF_STATS = 0;                        // 2048 doubles (4096 slots)
    const size_t OFF_MEAN  = 4096;                     // 1024
    const size_t OFF_RSTD  = 5120;                     // 1024
    const size_t OFF_GMAX  = 6144;                     // 8192
    const size_t OFF_IDX   = 14336;                    // 655360 ints
    const size_t OFF_XCAT  = OFF_IDX  + 655360;        // 8*192*4096
    const size_t OFF_A     = OFF_XCAT + (size_t)BATCH * 192 * NPTS;
    const size_t OFF_B     = OFF_A    + (size_t)BATCH * 64 * NK;

    double* statsd = (double*)(ws + OFF_STATS);
    float*  meanf  = ws + OFF_MEAN;
    float*  rstdf  = ws + OFF_RSTD;
    float*  gmax   = ws + OFF_GMAX;
    int*    idxb   = (int*)(ws + OFF_IDX);
    float*  xcat   = ws + OFF_XCAT;
    float*  A      = ws + OFF_A;
    float*  Bb     = ws + OFF_B;

    float* g    = A;                    // (b,1024,n)  33.5M floats < 41.9M
    float* cat2 = Bb;                   // (b,1216,n)  39.8M floats < 41.9M
    float* y7   = A;                    // (b,512,n)   16.8M floats
    float* y8   = A + 20971520;         // (b,256,n)    8.4M floats (disjoint from y7)

    const long bnkM = NK;               // 81920
    const long nM   = NPTS;             // 4096

    // ---- block 1: edgeconv(W1) -> edgeconv(W2) -> max_k -> x1 ----
    knn_kernel<3><<<dim3(NPTS, BATCH), 256, 0, stream>>>(x, 3L * NPTS, NPTS, idxb);
    edgeconv1_kernel<<<(BATCH * NPTS) / 8, 256, 0, stream>>>(x, idxb, W1, A);
    run_bn_mish(A, 64, bnkM, statsd, meanf, rstdf, stream);
    run_gemm(W2, A, Bb, 64, 64, bnkM, stream);
    run_bn_mish(Bb, 64, bnkM, statsd, meanf, rstdf, stream);
    {
        long tot = (long)BATCH * 64 * NPTS;
        max_k_kernel<<<(unsigned)((tot + 255) / 256), 256, 0, stream>>>(Bb, xcat, 0);
    }

    // ---- block 2: knn(x1) -> edgeconv(W3) -> conv(W4) -> max_k -> x2 ----
    knn_kernel<64><<<dim3(NPTS, BATCH), 256, 0, stream>>>(xcat, 192L * NPTS, NPTS, idxb);
    run_gemm_edge(W3, xcat, 192L * NPTS, NPTS, idxb, A, 64, 128, bnkM, stream);
    run_bn_mish(A, 64, bnkM, statsd, meanf, rstdf, stream);
    run_gemm(W4, A, Bb, 64, 64, bnkM, stream);
    run_bn_mish(Bb, 64, bnkM, statsd, meanf, rstdf, stream);
    {
        long tot = (long)BATCH * 64 * NPTS;
        max_k_kernel<<<(unsigned)((tot + 255) / 256), 256, 0, stream>>>(Bb, xcat, 64);
    }

    // ---- block 3: knn(x2) -> edgeconv(W5) -> max_k -> x3 ----
    knn_kernel<64><<<dim3(NPTS, BATCH), 256, 0, stream>>>(xcat + 64L * NPTS,
                                                          192L * NPTS, NPTS, idxb);
    run_gemm_edge(W5, xcat + 64L * NPTS, 192L * NPTS, NPTS, idxb, A, 64, 128, bnkM, stream);
    run_bn_mish(A, 64, bnkM, statsd, meanf, rstdf, stream);
    {
        long tot = (long)BATCH * 64 * NPTS;
        max_k_kernel<<<(unsigned)((tot + 255) / 256), 256, 0, stream>>>(A, xcat, 128);
    }

    // ---- global feature: conv(W6) -> bn+mish -> max over n -> broadcast concat ----
    run_gemm(W6, xcat, g, 1024, 192, nM, stream);
    run_bn_mish(g, 1024, nM, statsd, meanf, rstdf, stream);
    max_n_kernel<<<BATCH * 1024, 256, 0, stream>>>(g, gmax);
    {
        long tot = (long)BATCH * 1216 * NPTS;
        build_cat2<<<(unsigned)((tot + 255) / 256), 256, 0, stream>>>(gmax, xcat, cat2);
    }

    // ---- head: conv(W7) -> conv(W8) -> conv(W9) ----
    run_gemm(W7, cat2, y7, 512, 1216, nM, stream);
    run_bn_mish(y7, 512, nM, statsd, meanf, rstdf, stream);
    run_gemm(W8, y7, y8, 256, 512, nM, stream);
    run_bn_mish(y8, 256, nM, statsd, meanf, rstdf, stream);
    {
        long tot = (long)BATCH * 13 * NPTS;
        conv_last<<<(unsigned)((tot + 255) / 256), 256, 0, stream>>>(W9, y8, (float*)d_out);
    }
}